// GConv_65644280152909
// MI455X (gfx1250) — compile-verified
//
#include <hip/hip_runtime.h>
#include <hip/hip_bf16.h>

typedef __attribute__((ext_vector_type(2))) float v2f;
typedef __attribute__((ext_vector_type(8))) float v8f;

#define DH 256
#define OUTW 768
#define BN_EPS 1e-5f

__device__ __forceinline__ void atomAddF(float* p, float v) {
    unsafeAtomicAdd(p, v);   // native global_atomic_add_f32 on gfx1250
}

// ---------------------------------------------------------------- utilities
__global__ void zero_f(float* __restrict__ p, int n) {
    int stride = gridDim.x * blockDim.x;
    for (int i = blockIdx.x * blockDim.x + threadIdx.x; i < n; i += stride)
        p[i] = 0.0f;
}

__global__ void copy_f4(const float4* __restrict__ in, float4* __restrict__ out, int n4) {
    int stride = gridDim.x * blockDim.x;
    for (int i = blockIdx.x * blockDim.x + threadIdx.x; i < n4; i += stride)
        out[i] = in[i];
}

// ------------------------------------------------- edge scatter: h[dst] += z[src]
__global__ void scatter_add(const int* __restrict__ srcIdx, const int* __restrict__ dstIdx,
                            const float* __restrict__ z, float* __restrict__ h,
                            int E, int logd) {
    const int dmask = (1 << logd) - 1;
    const long long total = (long long)E << logd;
    const long long stride = (long long)gridDim.x * blockDim.x;
    for (long long i = (long long)blockIdx.x * blockDim.x + threadIdx.x; i < total; i += stride) {
        int e = (int)(i >> logd);
        int c = (int)(i & dmask);
        int s = srcIdx[e];
        int t = dstIdx[e];
        atomAddF(&h[((size_t)t << logd) + c], z[((size_t)s << logd) + c]);
    }
}

// ------------------------------------------------- WMMA fp32 GEMM: C = A[N,K] @ W[K,256] + bias
// One wave -> one 16x16 tile of C via V_WMMA_F32_16X16X4_F32.
__global__ void __launch_bounds__(256) gemm_wmma_f32(
    const float* __restrict__ A, const float* __restrict__ W,
    const float* __restrict__ bias, float* __restrict__ C,
    int Nrows, int K)
{
    const int lane = threadIdx.x & 31;
    const int wave = threadIdx.x >> 5;
    const int tile = blockIdx.x * 8 + wave;
    const int row0 = (tile >> 4) << 4;   // 16 column-tiles across DH=256
    const int col0 = (tile & 15) << 4;
    if (row0 >= Nrows) return;

    const int m    = lane & 15;          // M row (A) / N col (B,C)
    const int half = lane >> 4;          // selects K pair (A/B) and M+8 (C/D)

    v8f acc;
    const float b0 = bias[col0 + m];
#pragma unroll
    for (int v = 0; v < 8; ++v) acc[v] = b0;

    const float* arow = A + (size_t)(row0 + m) * K + 2 * half; // A[row0+m][k + 2*half]
    const float* wcol = W + (size_t)(2 * half) * DH + col0 + m; // W[k + 2*half][col0+m]

    for (int k = 0; k < K; k += 4) {
        v2f av;
        av[0] = arow[k];
        av[1] = arow[k + 1];
        v2f bv;
        bv[0] = wcol[(size_t)k * DH];
        bv[1] = wcol[(size_t)k * DH + DH];
        // D = A(16x4) x B(4x16) + C   -> v_wmma_f32_16x16x4_f32
        acc = __builtin_amdgcn_wmma_f32_16x16x4_f32(
                  false, av, false, bv, (short)0, acc, false, false);
    }

    float* crow = C + (size_t)(row0 + 8 * half) * DH + col0 + m;
#pragma unroll
    for (int v = 0; v < 8; ++v) crow[(size_t)v * DH] = acc[v];
}

// ------------------------------------------------- per-column sum / sum-of-squares
__global__ void __launch_bounds__(256) col_stats(
    const float* __restrict__ t, float* __restrict__ ssum, float* __restrict__ ssq,
    int N, int doRelu)
{
    const int c = threadIdx.x;           // 256 columns
    float s = 0.0f, q = 0.0f;
    for (int r = blockIdx.x; r < N; r += gridDim.x) {
        float v = t[(size_t)r * DH + c];
        if (doRelu) v = fmaxf(v, 0.0f);
        s += v;
        q += v * v;
    }
    atomAddF(&ssum[c], s);
    atomAddF(&ssq[c], q);
}

__global__ void bn_finalize(const float* __restrict__ ssum, const float* __restrict__ ssq,
                            const float* __restrict__ gamma, const float* __restrict__ beta,
                            float* __restrict__ scale, float* __restrict__ shift, int N)
{
    const int c = threadIdx.x;
    const float inv_n = 1.0f / (float)N;
    float mu  = ssum[c] * inv_n;
    float var = ssq[c] * inv_n - mu * mu;      // biased variance (torch BN)
    float s   = gamma[c] * rsqrtf(var + BN_EPS);
    scale[c] = s;
    shift[c] = beta[c] - mu * s;
}

// inner MLP: out = relu(BN(t))
__global__ void bn_relu(const float* __restrict__ t, const float* __restrict__ scale,
                        const float* __restrict__ shift, float* __restrict__ out, int total)
{
    int stride = gridDim.x * blockDim.x;
    for (int i = blockIdx.x * blockDim.x + threadIdx.x; i < total; i += stride) {
        int c = i & (DH - 1);
        out[i] = fmaxf(fmaf(t[i], scale[c], shift[c]), 0.0f);
    }
}

// outer: z = BN(relu(t2)); write z, out_z slice, atomic pool into out_g
__global__ void bn_pool_out(const float* __restrict__ t2, const float* __restrict__ scale,
                            const float* __restrict__ shift, const int* __restrict__ batch,
                            float* __restrict__ z, float* __restrict__ out_z,
                            float* __restrict__ out_g, int N, int layerOff)
{
    const int total = N * DH;
    int stride = gridDim.x * blockDim.x;
    for (int i = blockIdx.x * blockDim.x + threadIdx.x; i < total; i += stride) {
        int n = i >> 8;               // DH == 256
        int c = i & (DH - 1);
        float r = fmaxf(t2[i], 0.0f);
        float v = fmaf(r, scale[c], shift[c]);
        z[i] = v;
        out_z[(size_t)n * OUTW + layerOff + c] = v;
        atomAddF(&out_g[(size_t)batch[n] * OUTW + layerOff + c], v);
    }
}

// ---------------------------------------------------------------- driver
extern "C" void kernel_launch(void* const* d_in, const int* in_sizes, int n_in,
                              void* d_out, int out_size, void* d_ws, size_t ws_size,
                              hipStream_t stream)
{
    const float* x      = (const float*)d_in[0];
    const int*   ei     = (const int*)d_in[1];
    const int*   batch  = (const int*)d_in[2];

    const int N   = in_sizes[2];
    const int E   = in_sizes[1] / 2;
    const int DIN = in_sizes[0] / N;       // 128
    const int G   = out_size / OUTW - N;   // 256

    const int* srcIdx = ei;
    const int* dstIdx = ei + E;

    float* out_z = (float*)d_out;
    float* out_g = out_z + (size_t)N * OUTW;

    float* ws     = (float*)d_ws;
    float* buf_h  = ws;                          // N*256
    float* buf_t  = buf_h + (size_t)N * DH;      // N*256
    float* buf_z  = buf_t + (size_t)N * DH;      // N*256
    float* s_sum  = buf_z + (size_t)N * DH;      // 256
    float* s_sq   = s_sum + DH;                  // 256
    float* s_scale = s_sq + DH;                  // 256
    float* s_shift = s_scale + DH;               // 256

    // zero pooled-output region (atomically accumulated each call)
    {
        int n = G * OUTW;
        zero_f<<<(n + 255) / 256, 256, 0, stream>>>(out_g, n);
    }

    const int rowTiles = (N + 15) / 16;
    const int gemmBlocks = (rowTiles * 16 + 7) / 8;   // 8 waves (tiles) per block
    const int totNH = N * DH;
    const int ewBlocks = (totNH + 255) / 256;

    for (int l = 0; l < 3; ++l) {
        const float* W1  = (const float*)d_in[3 + 8 * l + 0];
        const float* b1  = (const float*)d_in[3 + 8 * l + 1];
        const float* g1  = (const float*)d_in[3 + 8 * l + 2];
        const float* be1 = (const float*)d_in[3 + 8 * l + 3];
        const float* W2  = (const float*)d_in[3 + 8 * l + 4];
        const float* b2  = (const float*)d_in[3 + 8 * l + 5];
        const float* go  = (const float*)d_in[3 + 8 * l + 6];
        const float* bo  = (const float*)d_in[3 + 8 * l + 7];

        const int din  = (l == 0) ? DIN : DH;
        const int logd = __builtin_ctz(din);
        const float* zin = (l == 0) ? x : buf_z;

        // h = z  (then scatter-add neighbors)
        int n4 = (N * din) / 4;
        copy_f4<<<(n4 + 255) / 256, 256, 0, stream>>>((const float4*)zin, (float4*)buf_h, n4);
        scatter_add<<<8192, 256, 0, stream>>>(srcIdx, dstIdx, zin, buf_h, E, logd);

        // t = h @ W1 + b1  (WMMA fp32)
        gemm_wmma_f32<<<gemmBlocks, 256, 0, stream>>>(buf_h, W1, b1, buf_t, N, din);

        // inner BN -> ReLU
        zero_f<<<2, 256, 0, stream>>>(s_sum, 2 * DH);
        col_stats<<<512, 256, 0, stream>>>(buf_t, s_sum, s_sq, N, 0);
        bn_finalize<<<1, 256, 0, stream>>>(s_sum, s_sq, g1, be1, s_scale, s_shift, N);
        bn_relu<<<ewBlocks, 256, 0, stream>>>(buf_t, s_scale, s_shift, buf_h, totNH);

        // t2 = h2 @ W2 + b2  (WMMA fp32)
        gemm_wmma_f32<<<gemmBlocks, 256, 0, stream>>>(buf_h, W2, b2, buf_t, N, DH);

        // outer: ReLU -> BN, emit z slice + pooled sums
        zero_f<<<2, 256, 0, stream>>>(s_sum, 2 * DH);
        col_stats<<<512, 256, 0, stream>>>(buf_t, s_sum, s_sq, N, 1);
        bn_finalize<<<1, 256, 0, stream>>>(s_sum, s_sq, go, bo, s_scale, s_shift, N);
        bn_pool_out<<<ewBlocks, 256, 0, stream>>>(buf_t, s_scale, s_shift, batch,
                                                  buf_z, out_z, out_g, N, l * DH);
    }
}